// Model_12945031430940
// MI455X (gfx1250) — compile-verified
//
#include <hip/hip_runtime.h>
#include <hip/hip_bf16.h>

// ---------------------------------------------------------------------------
// VQ-VAE forward for MI455X (gfx1250): conv1d layers as implicit GEMM on the
// fp32 WMMA pipe (V_WMMA_F32_16X16X4_F32). Weight + im2col activation chunks
// are staged into LDS so the hot loop is ds_load_b64 + v_wmma only.
// ---------------------------------------------------------------------------

#define B_   16
#define T_   4096
#define KCONV 5
#define NEG_SLOPE 0.02f
#define EPS_GN 1e-5f

typedef __attribute__((ext_vector_type(2))) float v2f;
typedef __attribute__((ext_vector_type(8))) float v8f;

// ---------------------------------------------------------------------------
// Implicit-GEMM conv1d (stride 1, 'same' padding) using fp32 WMMA 16x16x4.
//   Y[b, co, t] = bias[co] + sum_{ci,k} Wmat(co, ci, k) * Xcat[b, ci, t+k-PAD]
// Xcat = [X1 (Cin1 ch) ; X2 broadcast speaker cond (Cin2 ch, t-invariant)].
// FLIP=false: W is (Cout, Cin, KW)  [encoder / 1x1 mlp]
// FLIP=true : W is (Cin, Cout, KW), taps reversed [deconv stride 1]
//
// Block = 256 threads (8 waves); block tile = 16(co) x 128(t); each wave owns
// a 16x16 WMMA tile. K-dim (Cin*KW) walked in 16-channel chunks (CHUNK=16*KW):
//   wlds[m][r]  : weight chunk, r-major        -> A frag = one ds_load_b64
//   xlds[n][r]  : im2col activations, r-major  -> B frag = one ds_load_b64
// Row stride RS = CHUNK+2 (even): fragment bases are 8B aligned and the
// m*RS mod 64 bank pattern is distinct-even -> conflict-free b64 reads.
// ---------------------------------------------------------------------------
template <int KW, bool FLIP>
__global__ __launch_bounds__(256) void conv1d_wmma(
    const float* __restrict__ X1, const float* __restrict__ X2,
    const float* __restrict__ W, const float* __restrict__ bias,
    float* __restrict__ Y, int Cin1, int Cin2, int Cout)
{
  constexpr int PADW = (KW - 1) / 2;
  constexpr int TN = 128;            // time columns per block
  constexpr int CHUNK = 16 * KW;     // K-dim per chunk (80 or 16)
  constexpr int RS = CHUNK + 2;      // even row stride (dwords)

  __shared__ float wlds[16 * RS];    // A chunk: [co_local][r]
  __shared__ float xlds[TN * RS];    // B chunk: [n][r] (im2col)

  const int co0 = blockIdx.x * 16;
  const int t0 = blockIdx.y * TN;
  const int b = blockIdx.z;
  const int Cin = Cin1 + Cin2;

  const int tid = threadIdx.x;
  const int lane = tid & 31;
  const int wave = tid >> 5;
  const int half = lane >> 4;        // lane half -> K sub-pair (ks = 2*half)
  const int l16 = lane & 15;
  const int nloc = wave * 16 + l16;  // output time column within block

  // Fragment base pointers: index written as 2*(...) so clang proves 8B align
  // and emits single ds_load_b64 per fragment with immediate offsets.
  const v2f* pa = (const v2f*)&wlds[2 * (l16 * (RS / 2) + half)];
  const v2f* pb = (const v2f*)&xlds[2 * (nloc * (RS / 2) + half)];

  // im2col staging role: two threads per time column, each covers CHUNK/2 r's
  const int sn = tid >> 1;                  // column 0..127
  const int sr0 = (tid & 1) * (CHUNK / 2);  // starting r
  const int sci0 = sr0 / KW;                // starting channel within chunk
  const int tcol = t0 + sn - PADW;          // input time for k = 0

  v8f acc = {0.f, 0.f, 0.f, 0.f, 0.f, 0.f, 0.f, 0.f};

  for (int c0 = 0; c0 < Cin; c0 += 16) {
    __syncthreads();

    // ---- stage weight chunk: wlds[m][r], r = ci_local*KW + k ----
    for (int e = tid; e < 16 * CHUNK; e += 256) {
      int m = e / CHUNK, r = e % CHUNK;
      int ci = c0 + r / KW, k = r % KW;
      float wv;
      if (FLIP)
        wv = W[((size_t)ci * Cout + (co0 + m)) * KW + (KW - 1 - k)];
      else
        wv = W[((size_t)(co0 + m) * Cin + ci) * KW + k];
      wlds[m * RS + r] = wv;
    }

    // ---- stage im2col activations: xlds[n][r] = Xcat[c0+r/KW][tcol+r%KW] ----
    // (ci,k) tracked incrementally; chunk channels are uniformly X1 or X2
    // since every Cin1 in this net is a multiple of 16.
    {
      int ci = c0 + sci0;
      int k = sr0 % KW;  // 0 by construction (CHUNK/2 divisible by KW)
      const float* xrow = X1 + ((size_t)b * Cin1 + ci) * T_ + tcol;
#pragma unroll
      for (int j = 0; j < CHUNK / 2; ++j) {
        float v;
        if (ci < Cin1) {
          v = ((unsigned)(tcol + k) < (unsigned)T_) ? xrow[k] : 0.f;
        } else {
          v = X2[b * 128 + (ci - Cin1)];  // speaker cond: t-invariant
        }
        xlds[sn * RS + sr0 + j] = v;
        if (++k == KW) { k = 0; ++ci; xrow += T_; }
      }
    }

    // hint next weight chunk into cache (global_prefetch)
    if (c0 + 16 < Cin) {
      const float* wpf = FLIP ? &W[(size_t)(c0 + 16) * Cout * KW]
                              : &W[((size_t)co0 * Cin + (c0 + 16)) * KW];
      __builtin_prefetch(wpf, 0, 1);
    }
    __syncthreads();

    // ---- hot loop: 2x ds_load_b64 + fp32 WMMA per K-step of 4 ----
#pragma unroll
    for (int r = 0; r < CHUNK; r += 4) {
      v2f a = pa[r >> 1];
      v2f bf = pb[r >> 1];
      acc = __builtin_amdgcn_wmma_f32_16x16x4_f32(false, a, false, bf,
                                                  (short)0, acc, false, false);
    }
  }

  // ---- epilogue: C/D layout -> VGPR j holds M=j (+8 for upper lane half) ----
  const int n_out = t0 + nloc;
#pragma unroll
  for (int j = 0; j < 8; ++j) {
    int m_out = co0 + j + half * 8;
    Y[((size_t)b * Cout + m_out) * T_ + n_out] = acc[j] + bias[m_out];
  }
}

// ---------------------------------------------------------------------------
// Per-sample (C,T) normalization statistics: mean + rsqrt(var+eps).
// One block per batch sample, wave32 shuffle + LDS reduction.
// ---------------------------------------------------------------------------
__global__ void stats_kernel(const float* __restrict__ X, int n,
                             float* __restrict__ stats)
{
  const int b = blockIdx.x;
  const float* p = X + (size_t)b * n;
  float s = 0.f, ss = 0.f;
  for (int i = threadIdx.x; i < n; i += blockDim.x) {
    float v = p[i];
    s += v;
    ss += v * v;
  }
  for (int o = 16; o > 0; o >>= 1) {
    s += __shfl_down(s, o, 32);
    ss += __shfl_down(ss, o, 32);
  }
  __shared__ float sh0[8], sh1[8];
  const int wid = threadIdx.x >> 5;
  if ((threadIdx.x & 31) == 0) { sh0[wid] = s; sh1[wid] = ss; }
  __syncthreads();
  if (threadIdx.x == 0) {
    float S = 0.f, SS = 0.f;
    for (int i = 0; i < 8; ++i) { S += sh0[i]; SS += sh1[i]; }
    float m = S / (float)n;
    float var = SS / (float)n - m * m;
    stats[b * 2 + 0] = m;
    stats[b * 2 + 1] = rsqrtf(var + EPS_GN);
  }
}

// normalize + LeakyReLU (encoder), in place
__global__ void norm_lrelu_kernel(float* __restrict__ X,
                                  const float* __restrict__ g,
                                  const float* __restrict__ beta,
                                  const float* __restrict__ stats,
                                  int C, size_t total)
{
  size_t stride = (size_t)gridDim.x * blockDim.x;
  for (size_t i = (size_t)blockIdx.x * blockDim.x + threadIdx.x; i < total;
       i += stride) {
    int c = (int)((i / T_) % C);
    int b = (int)(i / ((size_t)T_ * C));
    float m = stats[b * 2 + 0], rs = stats[b * 2 + 1];
    float v = (X[i] - m) * rs * g[c] + beta[c];
    X[i] = (v >= 0.f) ? v : NEG_SLOPE * v;
  }
}

// normalize (over all 2*C channels) + GLU split (decoder): out has C channels
__global__ void norm_glu_kernel(const float* __restrict__ pre,
                                const float* __restrict__ g,
                                const float* __restrict__ beta,
                                const float* __restrict__ stats,
                                float* __restrict__ out, int C, size_t total)
{
  size_t stride = (size_t)gridDim.x * blockDim.x;
  for (size_t i = (size_t)blockIdx.x * blockDim.x + threadIdx.x; i < total;
       i += stride) {
    int t = (int)(i % T_);
    int c = (int)((i / T_) % C);
    int b = (int)(i / ((size_t)T_ * C));
    float m = stats[b * 2 + 0], rs = stats[b * 2 + 1];
    size_t base = ((size_t)b * (2 * C)) * T_;
    float a = pre[base + (size_t)c * T_ + t];
    float gg = pre[base + (size_t)(c + C) * T_ + t];
    float an = (a - m) * rs * g[c] + beta[c];
    float gn = (gg - m) * rs * g[c + C] + beta[c + C];
    out[i] = an * (1.f / (1.f + __expf(-gn)));
  }
}

// normalized speaker embedding: ye[b,:] = spk[y[b]] / (||.|| + 1e-8)
__global__ void ye_kernel(const float* __restrict__ spk,
                          const int* __restrict__ y, float* __restrict__ ye)
{
  const int b = blockIdx.x;
  const float* row = spk + (size_t)y[b] * 128;
  float v = row[threadIdx.x];
  float sq = v * v;
  for (int o = 16; o > 0; o >>= 1) sq += __shfl_down(sq, o, 32);
  __shared__ float sh[4];
  if ((threadIdx.x & 31) == 0) sh[threadIdx.x >> 5] = sq;
  __syncthreads();
  float nrm = sqrtf(sh[0] + sh[1] + sh[2] + sh[3]) + 1e-8f;
  ye[b * 128 + threadIdx.x] = v / nrm;
}

// row-normalized codebook
__global__ void cbn_kernel(const float* __restrict__ cb, float* __restrict__ cbn)
{
  int n = blockIdx.x * blockDim.x + threadIdx.x;
  if (n >= 512) return;
  const float* r = cb + (size_t)n * 64;
  float s = 0.f;
  for (int d = 0; d < 64; ++d) s += r[d] * r[d];
  float inv = 1.f / (sqrtf(s) + 1e-8f);
  for (int d = 0; d < 64; ++d) cbn[(size_t)n * 64 + d] = r[d] * inv;
}

// Fused VQ argmax: argmin ||zn - cbn||^2 == argmax z . cbn (cbn unit rows).
// Block handles 64 time positions; z tile staged in LDS; packed-key atomicMax.
__global__ void vq_argmax_kernel(const float* __restrict__ z,
                                 const float* __restrict__ cbn,
                                 int* __restrict__ idx)
{
  __shared__ float zt[64][65];
  __shared__ unsigned long long best[64];
  const int b = blockIdx.y;
  const int t0 = blockIdx.x * 64;

  for (int e = threadIdx.x; e < 64 * 64; e += blockDim.x) {
    int d = e >> 6, tt = e & 63;
    zt[d][tt] = z[((size_t)b * 64 + d) * T_ + t0 + tt];
  }
  if (threadIdx.x < 64) best[threadIdx.x] = 0ull;
  __syncthreads();

  for (int p = threadIdx.x; p < 512 * 64; p += blockDim.x) {
    int n = p >> 6, tt = p & 63;
    const float* c = cbn + (size_t)n * 64;
    float dot = 0.f;
#pragma unroll 8
    for (int d = 0; d < 64; ++d) dot = fmaf(c[d], zt[d][tt], dot);
    unsigned int fb = __float_as_uint(dot);
    fb = (fb & 0x80000000u) ? ~fb : (fb | 0x80000000u);  // monotonic order
    // tie-break toward the smallest index (matches argmin-first semantics)
    unsigned long long key =
        ((unsigned long long)fb << 32) | (unsigned int)(511 - n);
    atomicMax(&best[tt], key);
  }
  __syncthreads();
  if (threadIdx.x < 64)
    idx[(size_t)b * T_ + t0 + threadIdx.x] =
        511 - (int)(best[threadIdx.x] & 0xFFFFFFFFu);
}

// zq[b, d, t] = cbn[idx[b,t], d]
__global__ void vq_gather_kernel(const int* __restrict__ idx,
                                 const float* __restrict__ cbn,
                                 float* __restrict__ zq, size_t total)
{
  size_t stride = (size_t)gridDim.x * blockDim.x;
  for (size_t i = (size_t)blockIdx.x * blockDim.x + threadIdx.x; i < total;
       i += stride) {
    int t = (int)(i % T_);
    int d = (int)((i / T_) % 64);
    int b = (int)(i / ((size_t)T_ * 64));
    zq[i] = cbn[(size_t)idx[(size_t)b * T_ + t] * 64 + d];
  }
}

// ---------------------------------------------------------------------------
extern "C" void kernel_launch(void* const* d_in, const int* in_sizes, int n_in,
                              void* d_out, int out_size, void* d_ws,
                              size_t ws_size, hipStream_t stream)
{
  (void)in_sizes; (void)n_in; (void)out_size; (void)ws_size;

  const float* x   = (const float*)d_in[0];   // (B, 80, T)
  const int*   y   = (const int*)d_in[1];     // (B, 1)
  const float* spk = (const float*)d_in[2];   // (100, 128)
  const float* cb  = (const float*)d_in[3];   // (512, 64)
  // enc params: d_in[4 + 4l + {0:w,1:b,2:g,3:beta}]
  const float* mlp_w = (const float*)d_in[20]; // (64, 256, 1)
  const float* mlp_b = (const float*)d_in[21];
  // dec params: d_in[22 + 4l + ...] for l=0..2 ; d_in[34]=w, d_in[35]=b

  // ---- workspace carve-up ----
  char* w = (char*)d_ws;
  float* buf0 = (float*)w;  w += (size_t)B_ * 512 * T_ * sizeof(float);  // GEMM out
  float* buf1 = (float*)w;  w += (size_t)B_ * 512 * T_ * sizeof(float);  // ping-pong
  float* zbuf = (float*)w;  w += (size_t)B_ * 64 * T_ * sizeof(float);
  float* zq   = (float*)w;  w += (size_t)B_ * 64 * T_ * sizeof(float);
  float* yebuf = (float*)w; w += (size_t)B_ * 128 * sizeof(float);
  float* cbnbuf = (float*)w; w += 512 * 64 * sizeof(float);
  float* statsbuf = (float*)w; w += B_ * 2 * sizeof(float);
  int* idxbuf = (int*)w;    w += (size_t)B_ * T_ * sizeof(int);

  ye_kernel<<<B_, 128, 0, stream>>>(spk, y, yebuf);
  cbn_kernel<<<2, 256, 0, stream>>>(cb, cbnbuf);

  // ---- encoder: 4x [conv5 -> norm -> leaky relu] ----
  const float* src = x;
  int ci = 80;
  for (int l = 0; l < 4; ++l) {
    const float* Wl = (const float*)d_in[4 + 4 * l];
    const float* bl = (const float*)d_in[5 + 4 * l];
    const float* gl = (const float*)d_in[6 + 4 * l];
    const float* bt = (const float*)d_in[7 + 4 * l];
    float* dst = (l & 1) ? buf1 : buf0;
    conv1d_wmma<KCONV, false><<<dim3(256 / 16, T_ / 128, B_), 256, 0, stream>>>(
        src, nullptr, Wl, bl, dst, ci, 0, 256);
    stats_kernel<<<B_, 256, 0, stream>>>(dst, 256 * T_, statsbuf);
    norm_lrelu_kernel<<<2048, 256, 0, stream>>>(dst, gl, bt, statsbuf, 256,
                                                (size_t)B_ * 256 * T_);
    src = dst;
    ci = 256;
  }

  // ---- 1x1 conv to latent (z) ----
  conv1d_wmma<1, false><<<dim3(64 / 16, T_ / 128, B_), 256, 0, stream>>>(
      src, nullptr, mlp_w, mlp_b, zbuf, 256, 0, 64);

  // ---- vector quantization ----
  vq_argmax_kernel<<<dim3(T_ / 64, B_), 256, 0, stream>>>(zbuf, cbnbuf, idxbuf);
  vq_gather_kernel<<<2048, 256, 0, stream>>>(idxbuf, cbnbuf, zq,
                                             (size_t)B_ * 64 * T_);

  // ---- decoder: 3x [concat(ye) -> deconv5 -> norm -> GLU], then final ----
  float* postA = buf1;
  float* postB = buf1 + (size_t)B_ * 256 * T_;
  src = zq;
  int c1 = 64;
  for (int l = 0; l < 3; ++l) {
    const float* Wl = (const float*)d_in[22 + 4 * l];
    const float* bl = (const float*)d_in[23 + 4 * l];
    const float* gl = (const float*)d_in[24 + 4 * l];
    const float* bt = (const float*)d_in[25 + 4 * l];
    float* post = (l & 1) ? postB : postA;  // l0->A, l1->B, l2->A
    conv1d_wmma<KCONV, true><<<dim3(512 / 16, T_ / 128, B_), 256, 0, stream>>>(
        src, yebuf, Wl, bl, buf0, c1, 128, 512);
    stats_kernel<<<B_, 256, 0, stream>>>(buf0, 512 * T_, statsbuf);
    norm_glu_kernel<<<2048, 256, 0, stream>>>(buf0, gl, bt, statsbuf, post, 256,
                                              (size_t)B_ * 256 * T_);
    src = post;
    c1 = 256;
  }
  conv1d_wmma<KCONV, true><<<dim3(80 / 16, T_ / 128, B_), 256, 0, stream>>>(
      src, yebuf, (const float*)d_in[34], (const float*)d_in[35],
      (float*)d_out, 256, 128, 80);
}